// TwoWordPSDProbe_1580547973463
// MI455X (gfx1250) — compile-verified
//
#include <hip/hip_runtime.h>

// CDNA5 / gfx1250 — wave32, WMMA f32 16x16x4
typedef __attribute__((ext_vector_type(2))) float v2f;
typedef __attribute__((ext_vector_type(8))) float v8f;

#define MODEL_DIM 1024
#define RANK      256
#define BATCH     4
#define SEQ       512
#define ROWS      (BATCH * SEQ)   // 2048 flattened (b,s) rows

// -----------------------------------------------------------------------------
// Stage 1: T = batch @ proj      (2048 x 1024) * (1024 x 256) -> (2048 x 256)
// One wave per 64x16 output tile: 4 M-subtiles share one B fragment per K-step
// (B fragment = 2 stride-1KB scalar loads, the expensive one), 4 WMMAs/K-step.
//
// Per-lane fragment layout (ISA 7.12.2, 16x16x4 F32), lo=lane&15, hi=lane>>4:
//   A[v] = A[M = lo      ][K = 2*hi + v]   -> contiguous float2 from row (m+lo)
//   B[v] = B[K = 2*hi + v][N = lo      ]   -> column (n0+lo), stride RANK
//   C[v] = C[M = v + 8*hi][N = lo      ]
// -----------------------------------------------------------------------------
__global__ __launch_bounds__(256) void proj_gemm(const float* __restrict__ A,
                                                 const float* __restrict__ Bm,
                                                 float* __restrict__ T) {
  const int lane = threadIdx.x & 31;
  const int wave = threadIdx.x >> 5;
  const int wg   = blockIdx.x * 8 + wave;   // 0..511 tiles (32 Mx64  x  16 Nx16)
  const int mt   = wg >> 4;                 // 0..31  (M supertile of 64)
  const int nt   = wg & 15;                 // 0..15  (N tile of 16)
  const int lo   = lane & 15;
  const int hi   = lane >> 4;
  const int m0   = mt * 64;
  const int n0   = nt * 16;

  const float* __restrict__ a0 = A + (size_t)(m0 +  0 + lo) * MODEL_DIM;
  const float* __restrict__ a1 = A + (size_t)(m0 + 16 + lo) * MODEL_DIM;
  const float* __restrict__ a2 = A + (size_t)(m0 + 32 + lo) * MODEL_DIM;
  const float* __restrict__ a3 = A + (size_t)(m0 + 48 + lo) * MODEL_DIM;
  const float* __restrict__ bcol = Bm + (n0 + lo);

  v8f acc0 = {}, acc1 = {}, acc2 = {}, acc3 = {};
#pragma unroll 4
  for (int kk = 0; kk < MODEL_DIM; kk += 4) {
    const int k = kk + 2 * hi;
    v2f b;
    b.x = bcol[(size_t)(k + 0) * RANK];
    b.y = bcol[(size_t)(k + 1) * RANK];
    v2f fa0 = *(const v2f*)(a0 + k);   // 8B aligned: kk%4==0, 2*hi even
    v2f fa1 = *(const v2f*)(a1 + k);
    v2f fa2 = *(const v2f*)(a2 + k);
    v2f fa3 = *(const v2f*)(a3 + k);
    acc0 = __builtin_amdgcn_wmma_f32_16x16x4_f32(false, fa0, false, b, (short)0, acc0, false, false);
    acc1 = __builtin_amdgcn_wmma_f32_16x16x4_f32(false, fa1, false, b, (short)0, acc1, false, false);
    acc2 = __builtin_amdgcn_wmma_f32_16x16x4_f32(false, fa2, false, b, (short)0, acc2, false, false);
    acc3 = __builtin_amdgcn_wmma_f32_16x16x4_f32(false, fa3, false, b, (short)0, acc3, false, false);
  }

#pragma unroll
  for (int v = 0; v < 8; ++v) {
    const int m = v + 8 * hi;
    T[(size_t)(m0 +  0 + m) * RANK + n0 + lo] = acc0[v];
    T[(size_t)(m0 + 16 + m) * RANK + n0 + lo] = acc1[v];
    T[(size_t)(m0 + 32 + m) * RANK + n0 + lo] = acc2[v];
    T[(size_t)(m0 + 48 + m) * RANK + n0 + lo] = acc3[v];
  }
}

// -----------------------------------------------------------------------------
// Stage 1.5: norms[row] = ||T[row]||^2  — one wave32 per row, float4 loads,
// shfl_xor butterfly reduction.
// -----------------------------------------------------------------------------
__global__ __launch_bounds__(256) void row_norms(const float* __restrict__ T,
                                                 float* __restrict__ norms) {
  const int lane = threadIdx.x & 31;
  const int wave = threadIdx.x >> 5;
  const int row  = blockIdx.x * 8 + wave;   // 0..2047

  const float4* __restrict__ p = (const float4*)(T + (size_t)row * RANK) + lane;
  float4 x0 = p[0];     // floats [lane*4       .. lane*4+3]
  float4 x1 = p[32];    // floats [128 + lane*4 .. ]
  float s = x0.x * x0.x + x0.y * x0.y + x0.z * x0.z + x0.w * x0.w
          + x1.x * x1.x + x1.y * x1.y + x1.z * x1.z + x1.w * x1.w;

#pragma unroll
  for (int m = 16; m >= 1; m >>= 1)
    s += __shfl_xor(s, m, 32);

  if (lane == 0) norms[row] = s;
}

// -----------------------------------------------------------------------------
// Stage 2: out[b,i,j] = norms[i] + norms[j] - 2 * (T_i . T_j)
// One wave per 16x64 output tile: one A fragment (rows i0..i0+15 of T) shared
// across 4 j-tiles. B = T^T, which in fragment space is the identical
// contiguous-row float2 load pattern as A.
// -----------------------------------------------------------------------------
__global__ __launch_bounds__(256) void gram_dist(const float* __restrict__ T,
                                                 const float* __restrict__ norms,
                                                 float* __restrict__ out) {
  const int lane = threadIdx.x & 31;
  const int wave = threadIdx.x >> 5;
  const int wg   = blockIdx.x * 8 + wave;   // 0..1023  (4 b x 32 i x 8 j64)
  const int b    = wg >> 8;                 // 256 tiles per batch
  const int rem  = wg & 255;
  const int it   = rem >> 3;                // 0..31
  const int jt   = rem & 7;                 // 0..7
  const int lo   = lane & 15;
  const int hi   = lane >> 4;
  const int i0   = it * 16;
  const int j0   = jt * 64;

  const float* __restrict__ ti  = T + (size_t)(b * SEQ + i0 + lo) * RANK;
  const float* __restrict__ tj0 = T + (size_t)(b * SEQ + j0 +  0 + lo) * RANK;
  const float* __restrict__ tj1 = T + (size_t)(b * SEQ + j0 + 16 + lo) * RANK;
  const float* __restrict__ tj2 = T + (size_t)(b * SEQ + j0 + 32 + lo) * RANK;
  const float* __restrict__ tj3 = T + (size_t)(b * SEQ + j0 + 48 + lo) * RANK;

  v8f acc0 = {}, acc1 = {}, acc2 = {}, acc3 = {};
#pragma unroll 4
  for (int kk = 0; kk < RANK; kk += 4) {
    const int k = kk + 2 * hi;
    v2f a  = *(const v2f*)(ti  + k);
    v2f b0 = *(const v2f*)(tj0 + k);
    v2f b1 = *(const v2f*)(tj1 + k);
    v2f b2 = *(const v2f*)(tj2 + k);
    v2f b3 = *(const v2f*)(tj3 + k);
    acc0 = __builtin_amdgcn_wmma_f32_16x16x4_f32(false, a, false, b0, (short)0, acc0, false, false);
    acc1 = __builtin_amdgcn_wmma_f32_16x16x4_f32(false, a, false, b1, (short)0, acc1, false, false);
    acc2 = __builtin_amdgcn_wmma_f32_16x16x4_f32(false, a, false, b2, (short)0, acc2, false, false);
    acc3 = __builtin_amdgcn_wmma_f32_16x16x4_f32(false, a, false, b3, (short)0, acc3, false, false);
  }

  const float nj0 = norms[b * SEQ + j0 +  0 + lo];
  const float nj1 = norms[b * SEQ + j0 + 16 + lo];
  const float nj2 = norms[b * SEQ + j0 + 32 + lo];
  const float nj3 = norms[b * SEQ + j0 + 48 + lo];

#pragma unroll
  for (int v = 0; v < 8; ++v) {
    const int i    = i0 + v + 8 * hi;
    const float ni = norms[b * SEQ + i];
    float* __restrict__ orow = out + ((size_t)(b * SEQ + i)) * SEQ + j0 + lo;
    orow[ 0] = ni + nj0 - 2.0f * acc0[v];
    orow[16] = ni + nj1 - 2.0f * acc1[v];
    orow[32] = ni + nj2 - 2.0f * acc2[v];
    orow[48] = ni + nj3 - 2.0f * acc3[v];
  }
}

// -----------------------------------------------------------------------------
extern "C" void kernel_launch(void* const* d_in, const int* in_sizes, int n_in,
                              void* d_out, int out_size, void* d_ws, size_t ws_size,
                              hipStream_t stream) {
  const float* batch = (const float*)d_in[0];   // (4, 512, 1024) f32
  const float* proj  = (const float*)d_in[1];   // (1024, 256)    f32
  float* out   = (float*)d_out;                 // (4, 512, 512)  f32
  float* T     = (float*)d_ws;                  // 2048*256 f32 = 2 MB
  float* norms = T + (size_t)ROWS * RANK;       // 2048 f32

  // Stage 1: 512 waves (64x16 tiles), 8 waves/block -> 64 blocks
  proj_gemm<<<64, 256, 0, stream>>>(batch, proj, T);
  // Stage 1.5: 2048 rows, 8 waves/block -> 256 blocks
  row_norms<<<256, 256, 0, stream>>>(T, norms);
  // Stage 2: 1024 waves (16x64 tiles), 8 waves/block -> 128 blocks
  gram_dist<<<128, 256, 0, stream>>>(T, norms, out);
}